// CausalSE_16415365005630
// MI455X (gfx1250) — compile-verified
//
#include <hip/hip_runtime.h>
#include <math.h>

typedef __attribute__((ext_vector_type(2))) float v2f;
typedef __attribute__((ext_vector_type(4))) float v4f;
typedef __attribute__((ext_vector_type(8))) float v8f;

#define B_  8
#define D_  512
#define T_  8192
#define C_  512   // T / chunksize
#define DH_ 64
#define CH_ 16

// ---------------------------------------------------------------------------
// Kernel 1: per-(b,d) row -> 512 chunk sums -> inclusive scan -> pooled
// One 256-thread block per row (4096 rows). Row = 32KB, read as float4.
// ---------------------------------------------------------------------------
__global__ void cse_pooled(const float* __restrict__ x, float* __restrict__ pooled) {
  __shared__ float s[C_];
  const int row = blockIdx.x;                       // b*D + d
  const float* xr = x + (size_t)row * T_;
  const int t = threadIdx.x;                        // 0..255
#pragma unroll
  for (int i = 0; i < 2; ++i) {
    const int c = t + i * 256;
    const float4* p = (const float4*)(xr + c * CH_);
    float4 v0 = p[0], v1 = p[1], v2 = p[2], v3 = p[3];
    s[c] = ((v0.x + v0.y) + (v0.z + v0.w)) + ((v1.x + v1.y) + (v1.z + v1.w)) +
           ((v2.x + v2.y) + (v2.z + v2.w)) + ((v3.x + v3.y) + (v3.z + v3.w));
  }
  __syncthreads();
  // Hillis-Steele inclusive scan over 512 values, 2 elems/thread.
  // All reads complete before any write (barrier-separated) -> correct.
  for (int off = 1; off < C_; off <<= 1) {
    const float a0 = (t >= off) ? s[t - off] : 0.0f;
    const float a1 = s[t + 256 - off];               // t+256-off >= 0 always
    __syncthreads();
    s[t]       += a0;
    s[t + 256] += a1;
    __syncthreads();
  }
#pragma unroll
  for (int i = 0; i < 2; ++i) {
    const int c = t + i * 256;
    pooled[(size_t)row * C_ + c] = s[c] / (float)(CH_ * (c + 1));
  }
}

// ---------------------------------------------------------------------------
// fp32 WMMA helper: D = A(16x4) * B(4x16) + C, exact fp32 math.
// ---------------------------------------------------------------------------
__device__ __forceinline__ v8f wmma_f32x4(v2f a, v2f b, v8f c) {
  return __builtin_amdgcn_wmma_f32_16x16x4_f32(false, a, false, b, (short)0, c,
                                               false, false);
}

// ---------------------------------------------------------------------------
// Kernel 2: h[b,o,c] = relu(sum_d w1[o,d] * pooled[b,d,c] + b1[o])
// M=64 (4 tiles), N=512 (32 tiles), K=512. One 16x16 tile per wave.
// 1024 tiles total, 8 waves per 256-thread block -> 128 blocks.
// ---------------------------------------------------------------------------
__global__ void cse_gemm1(const float* __restrict__ w1, const float* __restrict__ b1,
                          const float* __restrict__ pooled, float* __restrict__ h) {
  const int lane = threadIdx.x & 31;
  const int tile = blockIdx.x * 8 + (threadIdx.x >> 5);   // 0..1023
  const int b  = tile >> 7;
  const int ot = (tile >> 5) & 3;
  const int ct = tile & 31;
  const int o0 = ot * 16, c0 = ct * 16;

  const int mn = lane & 15;                 // M for A-loads, N for B/C/D
  const int kb = (lane >> 4) << 1;          // lanes 16-31 carry K=2,3

  const float* Ar = w1 + (size_t)(o0 + mn) * D_;
  const float* Bb = pooled + (size_t)b * D_ * C_ + (c0 + mn);

  v8f acc = {};
  for (int k = 0; k < D_; k += 4) {
    v2f a, bb;
    a.x  = Ar[k + kb];
    a.y  = Ar[k + kb + 1];
    bb.x = Bb[(size_t)(k + kb) * C_];
    bb.y = Bb[(size_t)(k + kb + 1) * C_];
    acc = wmma_f32x4(a, bb, acc);
  }

  const int mo = (lane >> 4) * 8;           // C/D: lanes 16-31 hold M=r+8
#pragma unroll
  for (int r = 0; r < 8; ++r) {
    const int o = o0 + r + mo;
    const float v = acc[r] + b1[o];
    h[((size_t)b * DH_ + o) * C_ + c0 + mn] = fmaxf(v, 0.0f);
  }
}

// ---------------------------------------------------------------------------
// Kernel 3: g[b,d,c] = sigmoid(sum_o w2[d,o] * h[b,o,c] + b2[d])
// M=512 (32 tiles), N=512 (32 tiles), K=64. 8192 tiles -> 1024 blocks.
// ---------------------------------------------------------------------------
__global__ void cse_gemm2(const float* __restrict__ w2, const float* __restrict__ b2,
                          const float* __restrict__ h, float* __restrict__ g) {
  const int lane = threadIdx.x & 31;
  const int tile = blockIdx.x * 8 + (threadIdx.x >> 5);   // 0..8191
  const int b  = tile >> 10;
  const int mt = (tile >> 5) & 31;
  const int ct = tile & 31;
  const int d0 = mt * 16, c0 = ct * 16;

  const int mn = lane & 15;
  const int kb = (lane >> 4) << 1;

  const float* Ar = w2 + (size_t)(d0 + mn) * DH_;
  const float* Bb = h + (size_t)b * DH_ * C_ + (c0 + mn);

  v8f acc = {};
#pragma unroll
  for (int k = 0; k < DH_; k += 4) {
    v2f a, bb;
    a.x  = Ar[k + kb];
    a.y  = Ar[k + kb + 1];
    bb.x = Bb[(size_t)(k + kb) * C_];
    bb.y = Bb[(size_t)(k + kb + 1) * C_];
    acc = wmma_f32x4(a, bb, acc);
  }

  const int mo = (lane >> 4) * 8;
#pragma unroll
  for (int r = 0; r < 8; ++r) {
    const int d = d0 + r + mo;
    const float v = acc[r] + b2[d];
    g[((size_t)b * D_ + d) * C_ + c0 + mn] = 1.0f / (1.0f + __expf(-v));
  }
}

// ---------------------------------------------------------------------------
// Kernel 4: out = x * repeat(g,16). float4 vectorized (never crosses a chunk).
// x loads are regular (want L2 hits from pass 1); out stores are non-temporal
// so they do not evict x from the 192MB L2.
// ---------------------------------------------------------------------------
__global__ void cse_gate(const float* __restrict__ x, const float* __restrict__ g,
                         float* __restrict__ out) {
  const size_t idx = (size_t)blockIdx.x * 256 + threadIdx.x;  // float4 index
  const size_t row = idx >> 11;                               // / (T/4)
  const int    t4  = (int)(idx & 2047);
  const float  gg  = g[row * C_ + (t4 >> 2)];                 // chunk = (t4*4)/16
  v4f v = ((const v4f*)x)[idx];
  v *= gg;
  __builtin_nontemporal_store(v, (v4f*)out + idx);
}

// ---------------------------------------------------------------------------
extern "C" void kernel_launch(void* const* d_in, const int* in_sizes, int n_in,
                              void* d_out, int out_size, void* d_ws, size_t ws_size,
                              hipStream_t stream) {
  (void)in_sizes; (void)n_in; (void)out_size; (void)ws_size;
  const float* x  = (const float*)d_in[0];
  const float* w1 = (const float*)d_in[1];
  const float* b1 = (const float*)d_in[2];
  const float* w2 = (const float*)d_in[3];
  const float* b2 = (const float*)d_in[4];
  float* out = (float*)d_out;

  float* pooled = (float*)d_ws;                              // 8 MB
  float* h      = pooled + (size_t)B_ * D_ * C_;             // 1 MB
  float* g      = h + (size_t)B_ * DH_ * C_;                 // 8 MB

  cse_pooled<<<B_ * D_, 256, 0, stream>>>(x, pooled);
  cse_gemm1<<<(B_ * 4 * 32) / 8, 256, 0, stream>>>(w1, b1, pooled, h);
  cse_gemm2<<<(B_ * 32 * 32) / 8, 256, 0, stream>>>(w2, b2, h, g);
  cse_gate<<<(size_t)B_ * D_ * T_ / 4 / 256, 256, 0, stream>>>(x, g, out);
}